// RNN_Skills_Model_67405216744293
// MI455X (gfx1250) — compile-verified
//
#include <hip/hip_runtime.h>
#include <cstddef>

// ---------------------------------------------------------------------------
// Batch-1 8-layer LSTM, 256 sequential steps, MI455X (gfx1250).
// Weight-bandwidth-bound (88 MB fp32 weights/step, 11.3 GFLOP total).
// 88 MB < 192 MB L2 -> weights stay L2-resident; one persistent kernel with
// device-wide barriers covers the 10 sequential dependency points per step.
// GEMVs run as V_WMMA_F32_16X16X4_F32 tiles with broadcast-B (the 16x column
// redundancy is free under the 0.5 FLOP/byte roofline); weights are packed so
// each lane issues one global_load_b128 per TWO WMMA steps (512B/wave/iter).
// ---------------------------------------------------------------------------

#define DEV __device__ __forceinline__

typedef __attribute__((ext_vector_type(2))) float v2f;
typedef __attribute__((ext_vector_type(4))) float v4f;
typedef __attribute__((ext_vector_type(8))) float v8f;

constexpr int QN    = 4096;          // Q == N
constexpr int C     = 512;
constexpr int H     = 1024;
constexpr int L     = 8;
constexpr int G     = 4 * C;         // 2048 gates per layer
constexpr int STEPS = 256;

constexpr int NTHR  = 256;           // 8 wave32 per WG
constexpr int NWG   = 64;            // persistent grid (must co-reside)
constexpr int NWAVE = NWG * (NTHR / 32);   // 512 waves

// ---- packed weight layout (floats): per (tile, jpair, lane) a float4 ------
constexpr size_t SZ_WINP = (size_t)(H / 16) * (QN / 8) * 128;  // 64*512*128
constexpr size_t SZ_WIH0 = (size_t)(G / 16) * (H / 8) * 128;   // 128*128*128
constexpr size_t SZ_WL   = (size_t)(G / 16) * (C / 8) * 128;   // 128*64*128
constexpr size_t OFF_WINP = 0;
constexpr size_t OFF_WIH0 = OFF_WINP + SZ_WINP;
constexpr size_t OFF_WIHR = OFF_WIH0 + SZ_WIH0;                // 7 layers
constexpr size_t OFF_WHH  = OFF_WIHR + 7 * SZ_WL;              // 8 layers
constexpr size_t PACK_TOT = OFF_WHH + 8 * SZ_WL;               // ~22.0M floats

// ---- workspace layout (floats) --------------------------------------------
constexpr size_t OFF_X    = PACK_TOT;                 // current x (4096)
constexpr size_t OFF_ACCI = OFF_X + QN;               // inp accumulator (1024)
constexpr size_t OFF_ACCG = OFF_ACCI + H;             // gate accum (L*G)
constexpr size_t OFF_HB   = OFF_ACCG + (size_t)L * G; // h ping-pong (2*L*C)
constexpr size_t OFF_CB   = OFF_HB + 2 * (size_t)L * C;
constexpr size_t OFF_BAR  = OFF_CB + 2 * (size_t)L * C; // barrier words

DEV float sigm(float v) { return 1.0f / (1.0f + expf(-v)); }

// ---------------------------------------------------------------------------
// Device-wide sense-reversing barrier (all threads fence; thread 0 arrives).
// ---------------------------------------------------------------------------
DEV void grid_barrier(unsigned* cnt, unsigned* gen, unsigned& lg) {
  __threadfence();
  __syncthreads();
  if (threadIdx.x == 0) {
    unsigned g = lg;
    unsigned prev = atomicAdd(cnt, 1u);
    if (prev == (unsigned)(NWG - 1)) {
      *cnt = 0u;
      __threadfence();
      __hip_atomic_store(gen, g + 1u, __ATOMIC_RELEASE, __HIP_MEMORY_SCOPE_AGENT);
    } else {
      while (__hip_atomic_load(gen, __ATOMIC_ACQUIRE, __HIP_MEMORY_SCOPE_AGENT) == g) {
        __builtin_amdgcn_s_sleep(8);
      }
    }
  }
  lg++;
  __syncthreads();
}

// ---------------------------------------------------------------------------
// One GEMV K-chunk: 16 output rows (one tile), njp pair-steps (K=8 each).
// A: packed fp32 weights, one global_load_b128 per lane per pair-step
// (coalesced 512B/wave -> two WMMAs).  B: x-chunk broadcast from LDS into all
// 16 columns.  Result columns identical; lanes 0/16 hold rows 0-7 / 8-15 in
// the 8 accumulator VGPRs -> atomic-add into acc[].
// ---------------------------------------------------------------------------
DEV void gemv_chunk(const float* __restrict__ Wp, int J2total, int tile,
                    int jp0, int njp, const float* __restrict__ sxb, int lane,
                    float* __restrict__ acc) {
  v8f cacc = {};
  const int half = lane >> 4;  // 0: K 0..1, 1: K 2..3 within each WMMA
  const float* p = Wp + (((size_t)tile * J2total + jp0) * 32 + lane) * 4;
  const float* bx = sxb + 8 * jp0 + 2 * half;
  for (int blk = 0; blk < njp; blk += 4) {
    __builtin_prefetch(p + ((size_t)blk + 8) * 128, 0, 1);
#pragma unroll
    for (int u = 0; u < 4; ++u) {
      const int jp = blk + u;
      v4f a = *(const v4f*)(p + (size_t)jp * 128);
      v2f b0, b1;
      b0.x = bx[8 * jp + 0];
      b0.y = bx[8 * jp + 1];
      b1.x = bx[8 * jp + 4];
      b1.y = bx[8 * jp + 5];
      v2f a0; a0.x = a.x; a0.y = a.y;
      v2f a1; a1.x = a.z; a1.y = a.w;
      cacc = __builtin_amdgcn_wmma_f32_16x16x4_f32(
          false, a0, false, b0, (short)0, cacc, false, false);
      cacc = __builtin_amdgcn_wmma_f32_16x16x4_f32(
          false, a1, false, b1, (short)0, cacc, false, false);
    }
  }
  if ((lane & 15) == 0) {
    int rbase = tile * 16 + half * 8;
#pragma unroll
    for (int r = 0; r < 8; ++r) atomicAdd(&acc[rbase + r], cacc[r]);
  }
}

// ---------------------------------------------------------------------------
// Repack a (rows x K) row-major fp32 matrix into the paired WMMA A layout:
// P[((tile*J2 + jp)*32 + lane)*4 + {0,1,2,3}] =
//   W[row][8jp + 2half + {0,1}], W[row][8jp + 4 + 2half + {0,1}]
// ---------------------------------------------------------------------------
__global__ __launch_bounds__(NTHR) void pack_kernel(const float* __restrict__ W,
                                                    float* __restrict__ P,
                                                    int tiles, int K) {
  size_t idx = (size_t)blockIdx.x * blockDim.x + threadIdx.x;
  size_t total = (size_t)tiles * (K / 8) * 32;
  if (idx >= total) return;
  int lane = (int)(idx & 31);
  int J2 = K / 8;
  int jp = (int)((idx >> 5) % (size_t)J2);
  int tile = (int)((idx >> 5) / (size_t)J2);
  int row = tile * 16 + (lane & 15);
  int h2 = (lane >> 4) * 2;
  size_t rb = (size_t)row * K + 8 * jp + h2;
  P[4 * idx + 0] = W[rb + 0];
  P[4 * idx + 1] = W[rb + 1];
  P[4 * idx + 2] = W[rb + 4];
  P[4 * idx + 3] = W[rb + 5];
}

__global__ __launch_bounds__(NTHR) void init_kernel(
    const float* __restrict__ inputs, const float* __restrict__ b_inp,
    const float* __restrict__ b_ih, const float* __restrict__ b_hh,
    float* __restrict__ x, float* __restrict__ accI, float* __restrict__ accG,
    float* __restrict__ hbuf, float* __restrict__ cbuf,
    unsigned* __restrict__ bar) {
  int gt = blockIdx.x * blockDim.x + threadIdx.x;
  int nt = gridDim.x * blockDim.x;
  if (gt < 32) bar[gt] = 0u;
  for (int i = gt; i < QN; i += nt) x[i] = inputs[i];  // inputs row 0 only
  for (int i = gt; i < H; i += nt) accI[i] = b_inp[i];
  for (int i = gt; i < L * G; i += nt) accG[i] = b_ih[i] + b_hh[i];
  for (int i = gt; i < 2 * L * C; i += nt) { hbuf[i] = 0.f; cbuf[i] = 0.f; }
}

// ---------------------------------------------------------------------------
// Persistent kernel: 256 steps x 10 barrier phases.
// ---------------------------------------------------------------------------
__global__ __launch_bounds__(NTHR) void rnn_main(
    const float* __restrict__ Ppack, float* __restrict__ x,
    float* __restrict__ accI, float* __restrict__ accG,
    float* __restrict__ hbuf, float* __restrict__ cbuf,
    unsigned* __restrict__ bar, const float* __restrict__ b_inp,
    const float* __restrict__ b_ih, const float* __restrict__ b_hh,
    const float* __restrict__ Dm, const float* __restrict__ lvec,
    const int* __restrict__ cols, float* __restrict__ out) {
  __shared__ float sX[QN];  // 16 KB staging (x / inp+h / h)
  const int tid = threadIdx.x;
  const int wg = blockIdx.x;
  const int lane = tid & 31;
  const int wv = tid >> 5;
  const int w = wg * (NTHR / 32) + wv;  // global wave id [0, 512)
  const int gt = wg * NTHR + tid;       // global thread id [0, 16384)
  unsigned lg = 0;
  unsigned* cnt = bar;
  unsigned* gen = bar + 16;

  const float* P_winp = Ppack + OFF_WINP;
  const float* P_wih0 = Ppack + OFF_WIH0;
  const float* P_wihr = Ppack + OFF_WIHR;
  const float* P_whh  = Ppack + OFF_WHH;

  for (int s = 0; s < STEPS; ++s) {
    const int p = s & 1;
    const float* hprev = hbuf + (size_t)p * L * C;
    float* hnext       = hbuf + (size_t)(1 - p) * L * C;
    const float* cprev = cbuf + (size_t)p * L * C;
    float* cnext       = cbuf + (size_t)(1 - p) * L * C;

    // ---- Phase A: accI(+bias) += W_inp @ x ; also re-seed layer-7 gates ----
    for (int i = tid; i < QN; i += NTHR) sX[i] = x[i];
    if (gt < G) accG[(size_t)7 * G + gt] = b_ih[7 * G + gt] + b_hh[7 * G + gt];
    __syncthreads();
    gemv_chunk(P_winp, QN / 8, wg, wv * 64, 64, sX, lane, accI);
    grid_barrier(cnt, gen, lg);

    // ---- Layer 0: gates0 += W_ih0 @ inp + W_hh0 @ h_prev[0] ----
    for (int i = tid; i < H; i += NTHR) sX[i] = accI[i];
    for (int i = tid; i < C; i += NTHR) sX[H + i] = hprev[i];
    __syncthreads();
    for (int t = w; t < 6 * 128; t += NWAVE) {
      int tile = t & 127, ch = t >> 7;  // ch 0-3: W_ih0 (K=1024), 4-5: W_hh0
      if (ch < 4)
        gemv_chunk(P_wih0, H / 8, tile, ch * 32, 32, sX, lane, accG);
      else
        gemv_chunk(P_whh, C / 8, tile, (ch - 4) * 32, 32, sX + H, lane, accG);
    }
    grid_barrier(cnt, gen, lg);

    // ---- Layers 1..7: cell(l-1) in-WG, then gates_l accumulation ----
    for (int l = 1; l < L; ++l) {
      const float* gl = accG + (size_t)(l - 1) * G;
      for (int i = tid; i < C; i += NTHR) {
        float cn = sigm(gl[C + i]) * cprev[(l - 1) * C + i] +
                   sigm(gl[i]) * tanhf(gl[2 * C + i]);
        float hn = sigm(gl[3 * C + i]) * tanhf(cn);
        sX[i] = hn;
        if (wg == 0) { hnext[(l - 1) * C + i] = hn; cnext[(l - 1) * C + i] = cn; }
      }
      for (int i = tid; i < C; i += NTHR) sX[C + i] = hprev[l * C + i];
      __syncthreads();
      {
        int tile = w & 127, ch = w >> 7;  // exactly 512 tasks
        const float* Wp = (ch < 2) ? (P_wihr + (size_t)(l - 1) * SZ_WL)
                                   : (P_whh + (size_t)l * SZ_WL);
        const float* xb = (ch < 2) ? sX : (sX + C);
        gemv_chunk(Wp, C / 8, tile, (ch & 1) * 32, 32, xb, lane,
                   accG + (size_t)l * G);
      }
      grid_barrier(cnt, gen, lg);
    }

    // ---- Output: cell(7) -> skills; gather/exp head; feedback; re-seed ----
    {
      const float* gl = accG + (size_t)7 * G;
      for (int i = tid; i < C; i += NTHR) {
        float cn = sigm(gl[C + i]) * cprev[7 * C + i] +
                   sigm(gl[i]) * tanhf(gl[2 * C + i]);
        float hn = sigm(gl[3 * C + i]) * tanhf(cn);
        sX[i] = hn;
        if (wg == 0) { hnext[7 * C + i] = hn; cnext[7 * C + i] = cn; }
      }
      __syncthreads();
      if (gt < QN) {
        int col = cols[gt];
        float a = sX[col];
        float dv = Dm[(size_t)gt * C + col];
        float o = 1.0f - expf(-lvec[gt] * (a - dv));
        o = fmaxf(0.25f, o);
        out[(size_t)s * QN + gt] = o;
        x[gt] = o;  // feedback into next step
      }
      for (int i = gt; i < H; i += NWG * NTHR) accI[i] = b_inp[i];
      for (int i = gt; i < 7 * G; i += NWG * NTHR)
        accG[i] = b_ih[i] + b_hh[i];
      grid_barrier(cnt, gen, lg);
    }
  }

  // ---- tail: hN (L*C), cN (L*C), skills (C); final parity is 0 ----
  if (wg == 0) {
    const size_t ob = (size_t)STEPS * QN;
    for (int i = tid; i < L * C; i += NTHR) out[ob + i] = hbuf[i];
    for (int i = tid; i < L * C; i += NTHR) out[ob + L * C + i] = cbuf[i];
    for (int i = tid; i < C; i += NTHR) out[ob + 2 * L * C + i] = hbuf[7 * C + i];
  }
}

// ---------------------------------------------------------------------------
extern "C" void kernel_launch(void* const* d_in, const int* in_sizes, int n_in,
                              void* d_out, int out_size, void* d_ws,
                              size_t ws_size, hipStream_t stream) {
  const float* inputs = (const float*)d_in[0];
  const float* W_inp  = (const float*)d_in[1];
  const float* b_inp  = (const float*)d_in[2];
  const float* W_ih0  = (const float*)d_in[3];
  const float* W_ihr  = (const float*)d_in[4];
  const float* W_hh   = (const float*)d_in[5];
  const float* b_ih   = (const float*)d_in[6];
  const float* b_hh   = (const float*)d_in[7];
  const float* Dm     = (const float*)d_in[8];
  const float* lvec   = (const float*)d_in[9];
  const int*   cols   = (const int*)d_in[11];
  float* out = (float*)d_out;
  float* ws  = (float*)d_ws;

  auto launch_pack = [&](const float* W, float* dst, int rows, int K) {
    size_t quads = (size_t)rows * (K / 8) * 2;  // tiles*(K/8)*32
    int blocks = (int)((quads + NTHR - 1) / NTHR);
    pack_kernel<<<blocks, NTHR, 0, stream>>>(W, dst, rows / 16, K);
  };
  launch_pack(W_inp, ws + OFF_WINP, H, QN);
  launch_pack(W_ih0, ws + OFF_WIH0, G, H);
  for (int l = 1; l < L; ++l)
    launch_pack(W_ihr + (size_t)(l - 1) * G * C,
                ws + OFF_WIHR + (size_t)(l - 1) * SZ_WL, G, C);
  for (int l = 0; l < L; ++l)
    launch_pack(W_hh + (size_t)l * G * C, ws + OFF_WHH + (size_t)l * SZ_WL, G, C);

  init_kernel<<<NWG, NTHR, 0, stream>>>(inputs, b_inp, b_ih, b_hh, ws + OFF_X,
                                        ws + OFF_ACCI, ws + OFF_ACCG,
                                        ws + OFF_HB, ws + OFF_CB,
                                        (unsigned*)(ws + OFF_BAR));

  rnn_main<<<NWG, NTHR, 0, stream>>>(ws, ws + OFF_X, ws + OFF_ACCI,
                                     ws + OFF_ACCG, ws + OFF_HB, ws + OFF_CB,
                                     (unsigned*)(ws + OFF_BAR), b_inp, b_ih,
                                     b_hh, Dm, lvec, cols, out);
  (void)in_sizes; (void)n_in; (void)out_size; (void)ws_size;
}